// Mamba2Block_21904333209677
// MI455X (gfx1250) — compile-verified
//
#include <hip/hip_runtime.h>
#include <hip/hip_bf16.h>

// ---------------------------------------------------------------------------
// Mamba2 block on gfx1250:
//   1) batched LayerNorm (2048x1024)
//   2) fp32 WMMA GEMM: Xn @ w_in -> proj (u|z)        [v_wmma_f32_16x16x4_f32]
//   3) reset-aware depthwise conv (K=4) + SiLU -> uc
//   4) fp32 WMMA GEMM: uc @ w_x -> p (dt_raw|B|C)
//   5) 3-pass chunked associative scan over the SSM recurrence
//   6) fp32 WMMA GEMM: y @ w_out + b_out + x (residual) -> out
// All fp32 end-to-end (reference is fp32; weights ~42MB live in 192MB L2).
// ---------------------------------------------------------------------------

#define T_LEN   2048
#define DMODEL  1024
#define INNER   2048
#define STATE   64
#define CONVK   4
#define PROJW   (2*INNER)          // 4096
#define PW      (INNER + 2*STATE)  // 2176
#define EPSV    1e-05f
#define CHUNK   64
#define NCHUNK  32

typedef float v2f __attribute__((ext_vector_type(2)));
typedef float v8f __attribute__((ext_vector_type(8)));

// ---------------------------------------------------------------------------
// LayerNorm over each timestep row (1024 elems), 256 threads per row.
// ---------------------------------------------------------------------------
__global__ __launch_bounds__(256) void layernorm_kernel(
    const float* __restrict__ x, const float* __restrict__ scale,
    const float* __restrict__ bias, float* __restrict__ xn)
{
    __shared__ float red[256];
    const int t = blockIdx.x;
    const int tid = threadIdx.x;
    const float* xr = x + (size_t)t * DMODEL;

    float v[4];
    float s = 0.f;
#pragma unroll
    for (int i = 0; i < 4; ++i) { v[i] = xr[tid + 256 * i]; s += v[i]; }
    red[tid] = s; __syncthreads();
    for (int off = 128; off > 0; off >>= 1) {
        if (tid < off) red[tid] += red[tid + off];
        __syncthreads();
    }
    const float mu = red[0] * (1.f / DMODEL);
    __syncthreads();

    float sq = 0.f;
#pragma unroll
    for (int i = 0; i < 4; ++i) { float d = v[i] - mu; sq += d * d; }
    red[tid] = sq; __syncthreads();
    for (int off = 128; off > 0; off >>= 1) {
        if (tid < off) red[tid] += red[tid + off];
        __syncthreads();
    }
    const float rstd = rsqrtf(red[0] * (1.f / DMODEL) + EPSV);

    float* xo = xn + (size_t)t * DMODEL;
#pragma unroll
    for (int i = 0; i < 4; ++i) {
        int c = tid + 256 * i;
        xo[c] = (v[i] - mu) * rstd * scale[c] + bias[c];
    }
}

// ---------------------------------------------------------------------------
// fp32 WMMA GEMM: C[M x N] = A[M x K] * B[K x N] + bias[N] (+ residual).
// Wave tile 32x64: two 16-row M-subtiles share each B fragment (8 x v8f acc).
// Block = 8 waves (2 M x 4 N) => 64x256 block tile.
// k-loop unrolled 2x (k0 += 8): all 12 fragment loads are issued into
// independent registers before the 16 WMMAs, so load latency overlaps the
// matrix pipe instead of serializing wait->wmma per fragment.
// Ragged N handled branch-free: B columns clamped to N-1 in the k-loop
// (garbage lands in accumulator columns the epilogue never stores).
// A layout (16x4 tile): row = lane&15, K pair = k0 + 2*(lane>>4) + {0,1}.
// B layout (4x16 tile): col = lane&15, same K striping.
// C/D layout: VGPR i -> row = m0 + i + 8*(lane>>4), col = n0 + (lane&15).
// ---------------------------------------------------------------------------
__global__ __launch_bounds__(256) void gemm_wmma_f32(
    const float* __restrict__ A, const float* __restrict__ B,
    const float* __restrict__ bias, const float* __restrict__ residual,
    float* __restrict__ C, int M, int N, int Kdim,
    int lda, int ldb, int ldc, int ldres)
{
    const int lane = threadIdx.x & 31;
    const int wave = threadIdx.x >> 5;
    const int wm = wave & 1;
    const int wn = wave >> 1;
    const int m0 = blockIdx.y * 64 + wm * 32;
    const int n0 = blockIdx.x * 256 + wn * 64;
    const int half = lane >> 4;
    const int l15 = lane & 15;

    v8f acc0[4] = {};
    v8f acc1[4] = {};
    const float* arow0 = A + (size_t)(m0 + l15) * lda;
    const float* arow1 = arow0 + (size_t)16 * lda;

    int bcol[4];
#pragma unroll
    for (int j = 0; j < 4; ++j) {
        const int c = n0 + j * 16 + l15;
        bcol[j] = (c < N) ? c : (N - 1);
    }

    for (int k0 = 0; k0 < Kdim; k0 += 8) {
        v2f a0[2], a1[2], bb[2][4];
#pragma unroll
        for (int u = 0; u < 2; ++u) {
            const int kb = k0 + 4 * u + 2 * half;
            a0[u].x = arow0[kb]; a0[u].y = arow0[kb + 1];
            a1[u].x = arow1[kb]; a1[u].y = arow1[kb + 1];
            const float* brow = B + (size_t)kb * ldb;
#pragma unroll
            for (int j = 0; j < 4; ++j) {
                bb[u][j].x = brow[bcol[j]];
                bb[u][j].y = brow[ldb + bcol[j]];
            }
        }
#pragma unroll
        for (int u = 0; u < 2; ++u) {
#pragma unroll
            for (int j = 0; j < 4; ++j) {
                acc0[j] = __builtin_amdgcn_wmma_f32_16x16x4_f32(
                    false, a0[u], false, bb[u][j], (short)0, acc0[j], false, false);
                acc1[j] = __builtin_amdgcn_wmma_f32_16x16x4_f32(
                    false, a1[u], false, bb[u][j], (short)0, acc1[j], false, false);
            }
        }
    }

#pragma unroll
    for (int j = 0; j < 4; ++j) {
        const int col = n0 + j * 16 + l15;
        if (col < N) {
            const float bv = bias[col];
#pragma unroll
            for (int i = 0; i < 8; ++i) {
                const int row = m0 + half * 8 + i;
                float v0 = acc0[j][i] + bv;
                float v1 = acc1[j][i] + bv;
                if (residual) {
                    v0 += residual[(size_t)row * ldres + col];
                    v1 += residual[(size_t)(row + 16) * ldres + col];
                }
                C[(size_t)row * ldc + col] = v0;
                C[(size_t)(row + 16) * ldc + col] = v1;
            }
        }
    }
}

// ---------------------------------------------------------------------------
// Depthwise causal conv (K=4) with per-step reset masking + SiLU.
// ---------------------------------------------------------------------------
__global__ __launch_bounds__(256) void conv_silu_kernel(
    const float* __restrict__ proj, const float* __restrict__ init_conv,
    const int* __restrict__ done_seq, const float* __restrict__ conv_w,
    const float* __restrict__ conv_b, float* __restrict__ uc)
{
    const int d = blockIdx.x * blockDim.x + threadIdx.x;   // < INNER
    const int t = blockIdx.y;                              // < T

    float acc = conv_b[d] + conv_w[(CONVK - 1) * INNER + d] * proj[(size_t)t * PROJW + d];
    float gate = 1.f;
#pragma unroll
    for (int j = 1; j < CONVK; ++j) {
        const int s = t - j + 1;  // reset applied at step s masks tap u_{t-j}
        const float dp = (s <= 0) ? 0.f : (float)done_seq[s - 1];
        gate *= (1.f - dp);
        const int st = t - j;
        const float u = (st >= 0) ? proj[(size_t)st * PROJW + d]
                                  : init_conv[(CONVK - 1 + st) * INNER + d];
        acc += gate * conv_w[(CONVK - 1 - j) * INNER + d] * u;
    }
    uc[(size_t)t * INNER + d] = acc / (1.f + __expf(-acc));
}

// ---------------------------------------------------------------------------
// Scan pass A: per (chunk, channel) compute local end-state (zero init) and
// the chunk decay product.  Decay product is reconstructed analytically:
//   prod_t exp(dt_t*A[n])*reset_t = (prod_t reset_t) * exp(A[n] * sum_t dt_t)
// so no per-n running product is carried (saves 64 VGPRs + 64 VALU/step).
// ---------------------------------------------------------------------------
__global__ __launch_bounds__(64) void scan_chunk_kernel(
    const float* __restrict__ p, const float* __restrict__ uc,
    const int* __restrict__ done_seq, const float* __restrict__ A_log,
    const float* __restrict__ dt_bias,
    float* __restrict__ cstate, float* __restrict__ cdecay)
{
    __shared__ float Bsh[CHUNK * STATE];
    const int d = blockIdx.x * 64 + threadIdx.x;
    const int chunk = blockIdx.y;
    const int t0 = chunk * CHUNK;

    for (int i = threadIdx.x; i < CHUNK * STATE; i += 64) {
        const int tt = i >> 6, n = i & 63;
        Bsh[i] = p[(size_t)(t0 + tt) * PW + INNER + n];
    }
    __syncthreads();

    float Aa[STATE], s[STATE];
#pragma unroll
    for (int n = 0; n < STATE; ++n) {
        Aa[n] = -__expf(A_log[(size_t)d * STATE + n]);
        s[n] = 0.f;
    }
    const float dtb = dt_bias[d];
    float sumdt = 0.f;
    float rgate = 1.f;   // product of reset factors over the chunk (0 or 1)

    for (int tt = 0; tt < CHUNK; ++tt) {
        const int t = t0 + tt;
        const float reset = 1.f - ((t == 0) ? 0.f : (float)done_seq[t - 1]);
        const float dtr = p[(size_t)t * PW + d] + dtb;
        const float dt = (dtr > 20.f) ? dtr : log1pf(__expf(dtr));
        const float coef = dt * uc[(size_t)t * INNER + d];
        sumdt += dt;
        rgate *= reset;
#pragma unroll
        for (int n = 0; n < STATE; ++n) {
            const float f = __expf(dt * Aa[n]) * reset;
            s[n] = f * s[n] + coef * Bsh[tt * STATE + n];
        }
    }
    const size_t base = ((size_t)chunk * INNER + d) * STATE;
#pragma unroll
    for (int n = 0; n < STATE; ++n) {
        cstate[base + n] = s[n];
        cdecay[base + n] = rgate * __expf(Aa[n] * sumdt);
    }
}

// ---------------------------------------------------------------------------
// Scan pass B: sequential combine across the 32 chunks per (d, n) pair;
// rewrites cstate[chunk] with the *initial* state entering that chunk.
// ---------------------------------------------------------------------------
__global__ __launch_bounds__(256) void scan_combine_kernel(
    const float* __restrict__ init_ssm, float* __restrict__ cstate,
    const float* __restrict__ cdecay)
{
    const int idx = blockIdx.x * 256 + threadIdx.x;  // < INNER*STATE
    float run = init_ssm[idx];
    for (int c = 0; c < NCHUNK; ++c) {
        const size_t off = (size_t)c * INNER * STATE + idx;
        const float st = cstate[off];
        cstate[off] = run;
        run = cdecay[off] * run + st;
    }
}

// ---------------------------------------------------------------------------
// Scan pass C: replay each chunk with the correct initial state, contract
// against C_t, add D*uc, gate with silu(z).
// ---------------------------------------------------------------------------
__global__ __launch_bounds__(64) void scan_apply_kernel(
    const float* __restrict__ p, const float* __restrict__ uc,
    const float* __restrict__ proj, const int* __restrict__ done_seq,
    const float* __restrict__ A_log, const float* __restrict__ dt_bias,
    const float* __restrict__ Dvec, const float* __restrict__ cstate,
    float* __restrict__ y)
{
    __shared__ float Bsh[CHUNK * STATE];
    __shared__ float Csh[CHUNK * STATE];
    const int d = blockIdx.x * 64 + threadIdx.x;
    const int chunk = blockIdx.y;
    const int t0 = chunk * CHUNK;

    for (int i = threadIdx.x; i < CHUNK * STATE; i += 64) {
        const int tt = i >> 6, n = i & 63;
        Bsh[i] = p[(size_t)(t0 + tt) * PW + INNER + n];
        Csh[i] = p[(size_t)(t0 + tt) * PW + INNER + STATE + n];
    }
    __syncthreads();

    float Aa[STATE], s[STATE];
    const size_t base = ((size_t)chunk * INNER + d) * STATE;
#pragma unroll
    for (int n = 0; n < STATE; ++n) {
        Aa[n] = -__expf(A_log[(size_t)d * STATE + n]);
        s[n] = cstate[base + n];
    }
    const float dtb = dt_bias[d];
    const float Dd = Dvec[d];

    for (int tt = 0; tt < CHUNK; ++tt) {
        const int t = t0 + tt;
        const float reset = 1.f - ((t == 0) ? 0.f : (float)done_seq[t - 1]);
        const float dtr = p[(size_t)t * PW + d] + dtb;
        const float dt = (dtr > 20.f) ? dtr : log1pf(__expf(dtr));
        const float uv = uc[(size_t)t * INNER + d];
        const float coef = dt * uv;
        float ysum = 0.f;
#pragma unroll
        for (int n = 0; n < STATE; ++n) {
            const float f = __expf(dt * Aa[n]) * reset;
            s[n] = f * s[n] + coef * Bsh[tt * STATE + n];
            ysum += s[n] * Csh[tt * STATE + n];
        }
        float yv = ysum + Dd * uv;
        const float z = proj[(size_t)t * PROJW + INNER + d];
        yv *= z / (1.f + __expf(-z));
        y[(size_t)t * INNER + d] = yv;
    }
}

// ---------------------------------------------------------------------------
extern "C" void kernel_launch(void* const* d_in, const int* in_sizes, int n_in,
                              void* d_out, int out_size, void* d_ws, size_t ws_size,
                              hipStream_t stream) {
    const float* x_seq      = (const float*)d_in[0];
    const int*   done_seq   = (const int*)  d_in[1];
    const float* init_ssm   = (const float*)d_in[2];
    const float* init_conv  = (const float*)d_in[3];
    const float* norm_scale = (const float*)d_in[4];
    const float* norm_bias  = (const float*)d_in[5];
    const float* w_in       = (const float*)d_in[6];
    const float* b_in       = (const float*)d_in[7];
    const float* w_x        = (const float*)d_in[8];
    const float* b_x        = (const float*)d_in[9];
    const float* w_out      = (const float*)d_in[10];
    const float* b_out      = (const float*)d_in[11];
    const float* conv_w     = (const float*)d_in[12];
    const float* conv_b     = (const float*)d_in[13];
    const float* A_log      = (const float*)d_in[14];
    const float* Dvec       = (const float*)d_in[15];
    const float* dt_bias    = (const float*)d_in[16];
    float* out = (float*)d_out;

    // Workspace carve-up (all fp32).
    float* ws     = (float*)d_ws;
    float* xn     = ws;                                    // T*DMODEL
    float* proj   = xn     + (size_t)T_LEN * DMODEL;       // T*PROJW
    float* uc     = proj   + (size_t)T_LEN * PROJW;        // T*INNER
    float* pbuf   = uc     + (size_t)T_LEN * INNER;        // T*PW
    float* ybuf   = pbuf   + (size_t)T_LEN * PW;           // T*INNER
    float* cstate = ybuf   + (size_t)T_LEN * INNER;        // NCHUNK*INNER*STATE
    float* cdecay = cstate + (size_t)NCHUNK * INNER * STATE;

    // 1) LayerNorm all timesteps.
    layernorm_kernel<<<T_LEN, 256, 0, stream>>>(x_seq, norm_scale, norm_bias, xn);

    // 2) proj = xn @ w_in + b_in   (2048x1024 @ 1024x4096)
    gemm_wmma_f32<<<dim3(PROJW / 256, T_LEN / 64), 256, 0, stream>>>(
        xn, w_in, b_in, nullptr, proj,
        T_LEN, PROJW, DMODEL, DMODEL, PROJW, PROJW, 0);

    // 3) uc = silu(conv(u) + conv_b) with reset masking.
    conv_silu_kernel<<<dim3(INNER / 256, T_LEN), 256, 0, stream>>>(
        proj, init_conv, done_seq, conv_w, conv_b, uc);

    // 4) p = uc @ w_x + b_x        (2048x2048 @ 2048x2176, N-tail clamped)
    gemm_wmma_f32<<<dim3((PW + 255) / 256, T_LEN / 64), 256, 0, stream>>>(
        uc, w_x, b_x, nullptr, pbuf,
        T_LEN, PW, INNER, INNER, PW, PW, 0);

    // 5) chunked associative scan over the SSM recurrence.
    scan_chunk_kernel<<<dim3(INNER / 64, NCHUNK), 64, 0, stream>>>(
        pbuf, uc, done_seq, A_log, dt_bias, cstate, cdecay);
    scan_combine_kernel<<<(INNER * STATE) / 256, 256, 0, stream>>>(
        init_ssm, cstate, cdecay);
    scan_apply_kernel<<<dim3(INNER / 64, NCHUNK), 64, 0, stream>>>(
        pbuf, uc, proj, done_seq, A_log, dt_bias, Dvec, cstate, ybuf);

    // 6) out = y @ w_out + b_out + x_seq   (2048x2048 @ 2048x1024)
    gemm_wmma_f32<<<dim3(DMODEL / 256, T_LEN / 64), 256, 0, stream>>>(
        ybuf, w_out, b_out, x_seq, out,
        T_LEN, DMODEL, INNER, INNER, DMODEL, DMODEL, DMODEL);
}